// PropagLayer_15092515078206
// MI455X (gfx1250) — compile-verified
//
#include <hip/hip_runtime.h>

typedef __attribute__((ext_vector_type(2))) float v2f;
typedef __attribute__((ext_vector_type(8))) float v8f;

#define WAVES_PER_BLOCK 8
#define THREADS_PER_BLOCK (WAVES_PER_BLOCK * 32)
#define TROWS 16            // time rows per block == WMMA M
#define MODES_PER_WAVE 8    // 2 WMMA K-groups of 4 per iteration

__global__ __launch_bounds__(THREADS_PER_BLOCK)
void propag_wmma_kernel(const float* __restrict__ params,
                        const float* __restrict__ time_grid,
                        float* __restrict__ out,
                        int n_modes)
{
    const float* __restrict__ Gp = params;                 // Gamma (coupling)
    const float* __restrict__ Wp = params + n_modes;       // omega (frequency)
    const float* __restrict__ Dp = params + 2 * n_modes;   // gamma (decay)

    const int lane = threadIdx.x & 31;
    const int wave = threadIdx.x >> 5;
    const int m    = lane & 15;    // time row 0..15  (A-matrix M index)
    const int half = lane >> 4;    // 0 -> K pair {0,1}; 1 -> K pair {2,3}
    const int t0   = blockIdx.x * TROWS;

    const float t = time_grid[t0 + m];

    // Two independent accumulator chains per component to break the serial
    // WMMA C->C dependency (merged after the loop).
    v8f cre0 = {}, cim0 = {};
    v8f cre1 = {}, cim1 = {};
    const v2f bones = {1.0f, 1.0f};

    const int stride = WAVES_PER_BLOCK * MODES_PER_WAVE;   // 64 modes / block iter

    for (int k0 = wave * MODES_PER_WAVE; k0 < n_modes; k0 += stride) {
        const int ka = k0 + half * 2;          // group A: modes ka, ka+1
        const int kbq = k0 + 4 + half * 2;     // group B: modes kbq, kbq+1

        // prefetch the next strip of the parameter stream (speculative OK)
        __builtin_prefetch(Gp + ka + stride, 0, 0);

        // ---- issue all loads for both groups up front ----
        float Ga0 = Gp[ka],      Ga1 = Gp[ka + 1];
        float wa0 = Wp[ka],      wa1 = Wp[ka + 1];
        float da0 = Dp[ka],      da1 = Dp[ka + 1];
        float Gb0 = Gp[kbq],     Gb1 = Gp[kbq + 1];
        float wb0 = Wp[kbq],     wb1 = Wp[kbq + 1];
        float db0 = Dp[kbq],     db1 = Dp[kbq + 1];

        // ---- group A ----
        float ea0 = __expf(-da0 * t);
        float ea1 = __expf(-da1 * t);
        float sa0, ca0, sa1, ca1;
        __sincosf(wa0 * t, &sa0, &ca0);
        __sincosf(wa1 * t, &sa1, &ca1);
        float aa0 = Ga0 * ea0;
        float aa1 = Ga1 * ea1;
        v2f Are0 = { aa0 * ca0, aa1 * ca1 };
        v2f Aim0 = { aa0 * sa0, aa1 * sa1 };

        // ---- group B ----
        float eb0 = __expf(-db0 * t);
        float eb1 = __expf(-db1 * t);
        float sb0, cb0, sb1, cb1;
        __sincosf(wb0 * t, &sb0, &cb0);
        __sincosf(wb1 * t, &sb1, &cb1);
        float ab0 = Gb0 * eb0;
        float ab1 = Gb1 * eb1;
        v2f Are1 = { ab0 * cb0, ab1 * cb1 };
        v2f Aim1 = { ab0 * sb0, ab1 * sb1 };

        // D = A x ones + C : per-time-row mode sums, 4 independent chains
        cre0 = __builtin_amdgcn_wmma_f32_16x16x4_f32(
                   false, Are0, false, bones, (short)0, cre0, false, false);
        cim0 = __builtin_amdgcn_wmma_f32_16x16x4_f32(
                   false, Aim0, false, bones, (short)0, cim0, false, false);
        cre1 = __builtin_amdgcn_wmma_f32_16x16x4_f32(
                   false, Are1, false, bones, (short)0, cre1, false, false);
        cim1 = __builtin_amdgcn_wmma_f32_16x16x4_f32(
                   false, Aim1, false, bones, (short)0, cim1, false, false);
    }

    // merge the two accumulator chains
    v8f cre = cre0 + cre1;
    v8f cim = cim0 + cim1;

    // C/D layout: VGPR r -> (lanes 0-15: M=r, N=lane) (lanes 16-31: M=r+8).
    // Every column N holds the same row sum, so lane 0 carries rows 0..7 and
    // lane 16 carries rows 8..15 in VGPRs 0..7.
    __shared__ float red_re[WAVES_PER_BLOCK][TROWS];
    __shared__ float red_im[WAVES_PER_BLOCK][TROWS];

    if (lane == 0 || lane == 16) {
        const int base = half * 8;
        #pragma unroll
        for (int r = 0; r < 8; ++r) {
            red_re[wave][base + r] = cre[r];
            red_im[wave][base + r] = cim[r];
        }
    }
    __syncthreads();

    if (threadIdx.x < TROWS) {
        float sre = 0.0f, sim = 0.0f;
        #pragma unroll
        for (int w = 0; w < WAVES_PER_BLOCK; ++w) {
            sre += red_re[w][threadIdx.x];
            sim += red_im[w][threadIdx.x];
        }
        // complex64 output: interleaved (Re, Im) float pairs
        float2 v = make_float2(sre, sim);
        *reinterpret_cast<float2*>(out + 2 * (t0 + threadIdx.x)) = v;
    }
}

extern "C" void kernel_launch(void* const* d_in, const int* in_sizes, int n_in,
                              void* d_out, int out_size, void* d_ws, size_t ws_size,
                              hipStream_t stream) {
    const float* params    = (const float*)d_in[0];  // [3, N] float32
    const float* time_grid = (const float*)d_in[1];  // [T]    float32
    float*       out       = (float*)d_out;          // [T] complex64 -> 2T floats

    const int n_modes = in_sizes[0] / 3;
    const int T       = in_sizes[1];
    const int blocks  = T / TROWS;                   // 16384/16 = 1024 blocks

    propag_wmma_kernel<<<blocks, THREADS_PER_BLOCK, 0, stream>>>(
        params, time_grid, out, n_modes);
}